// LorentzKernelTransform_21199958573234
// MI455X (gfx1250) — compile-verified
//
#include <hip/hip_runtime.h>
#include <math.h>

typedef __attribute__((ext_vector_type(2))) float v2f;
typedef __attribute__((ext_vector_type(8))) float v8f;

#define TILES_N 8   // 8 x 16 = 128 output channels
#define K_STEPS 16  // 16 x 4  = 64 input channels
#define HH 64
#define WW 64
#define CIN 64
#define COUT 128
#define KL 9

// One wave32 computes a 16-row x 128-col tile of y = P * W^T + b and the
// full Lorentz epilogue, entirely in registers.
__global__ __launch_bounds__(256) void lorentz_kernel_transform(
    const float* __restrict__ x,        // (16,64,64,64)
    const float* __restrict__ Wt,       // (128,64)
    const float* __restrict__ bias,     // (128)
    const float* __restrict__ kernels,  // (9,128)
    float* __restrict__ out)            // (16,4096,9,128)
{
  const int lane   = threadIdx.x & 31;
  const int wave   = threadIdx.x >> 5;
  const int lane15 = lane & 15;
  const int laneHi = lane >> 4;  // 0: K/M-rows 0..7 half, 1: other half

  const int tile    = blockIdx.x * 8 + wave;  // 16-row M tile index
  const int rowBase = tile * 16;

  // ---- A-fragment row for this lane: row = rowBase + lane15 (both halves
  // load the same row, different K pairs per the 16x4 f32 A layout) ----
  const float* rowptr;
  bool inb;
  {
    int r   = rowBase + lane15;       // global (pixel, kpos) row
    int k   = r % KL;                 // kernel position 0..8
    int pix = r / KL;                 // b*4096 + h*64 + w
    int b   = pix >> 12;
    int l   = pix & 4095;
    int h   = l >> 6;
    int w   = l & 63;
    int hh  = h + (k / 3) - 1;        // PAD = 1
    int ww  = w + (k % 3) - 1;
    inb = (hh >= 0) & (hh < HH) & (ww >= 0) & (ww < WW);
    int hs = inb ? hh : 0;
    int ws = inb ? ww : 0;
    rowptr = x + ((size_t)((b * HH + hs) * WW + ws)) * CIN;
  }

  // ---- Load all 16 A fragments (float2 per lane per K-chunk) ----
  v2f a[K_STEPS];
#pragma unroll
  for (int s = 0; s < K_STEPS; ++s) {
    int c0   = 4 * s + 2 * laneHi;
    float a0 = rowptr[c0];
    float a1 = rowptr[c0 + 1];
    a0 = inb ? a0 : 0.0f;
    a1 = inb ? a1 : 0.0f;
    if (c0 == 0) a0 = fmaxf(a0, 1.0f);  // p_time clamp (pads become 1.0 too)
    a[s].x = a0;
    a[s].y = a1;
  }

  // ---- 128x v_wmma_f32_16x16x4_f32 : s-outer / t-inner for ILP ----
  v8f acc[TILES_N];
#pragma unroll
  for (int t = 0; t < TILES_N; ++t) acc[t] = (v8f){};

  const float* wl = Wt + (size_t)lane15 * CIN + 2 * laneHi;  // col base per lane
#pragma unroll
  for (int s = 0; s < K_STEPS; ++s) {
#pragma unroll
    for (int t = 0; t < TILES_N; ++t) {
      // B element pair: column o = 16*t + lane15, K rows (4s+2*laneHi, +1)
      v2f bf = *(const v2f*)(wl + (size_t)(16 * t) * CIN + 4 * s);
      acc[t] = __builtin_amdgcn_wmma_f32_16x16x4_f32(
          false, a[s], false, bf, (short)0, acc[t], false, false);
    }
  }

  // ---- Epilogue: bias add + per-row sum of squares (space dims only) ----
  float part[8];
#pragma unroll
  for (int i = 0; i < 8; ++i) part[i] = 0.0f;

#pragma unroll
  for (int t = 0; t < TILES_N; ++t) {
    float bt = bias[16 * t + lane15];
#pragma unroll
    for (int i = 0; i < 8; ++i) {
      float y   = acc[t][i] + bt;
      acc[t][i] = y;
      float yy  = y * y;
      if (t == 0) yy = (lane15 == 0) ? 0.0f : yy;  // exclude time channel n==0
      part[i] += yy;
    }
  }

  // Half-wave (16-lane) reduction: masks 1,2,4,8 stay within each half,
  // matching the D layout (lanes 0-15 hold rows 0-7, lanes 16-31 rows 8-15).
#pragma unroll
  for (int i = 0; i < 8; ++i) {
    part[i] += __shfl_xor(part[i], 1, 32);
    part[i] += __shfl_xor(part[i], 2, 32);
    part[i] += __shfl_xor(part[i], 4, 32);
    part[i] += __shfl_xor(part[i], 8, 32);
  }

  float scale[8];
#pragma unroll
  for (int i = 0; i < 8; ++i) {
    float s2    = part[i];
    float yt    = sqrtf(s2 + 1.0f);             // K = 1
    float alpha = fmaxf(yt, 1.0f + 1e-7f);      // EPS
    float dist  = acoshf(alpha);                // SQRT_K = 1
    float sn    = sqrtf(s2);
    scale[i]    = dist / fmaxf(sn, 1e-8f);
  }

  // ---- Store: out[row, n] = scale*y - kernels[k(row), n]; n==0 -> -kernel ----
#pragma unroll
  for (int i = 0; i < 8; ++i) {
    int m  = i + 8 * laneHi;
    int r  = rowBase + m;
    int km = r % KL;
    size_t obase    = (size_t)r * COUT;
    const float* kr = kernels + km * COUT;
#pragma unroll
    for (int t = 0; t < TILES_N; ++t) {
      int n    = 16 * t + lane15;
      float kv = kr[n];
      float ov = scale[i] * acc[t][i] - kv;
      if (t == 0) ov = (lane15 == 0) ? (0.0f - kv) : ov;
      out[obase + n] = ov;
    }
  }
}

extern "C" void kernel_launch(void* const* d_in, const int* in_sizes, int n_in,
                              void* d_out, int out_size, void* d_ws, size_t ws_size,
                              hipStream_t stream) {
  const float* x       = (const float*)d_in[0];  // (16,64,64,64)
  const float* wt      = (const float*)d_in[1];  // (128,64)
  const float* bias    = (const float*)d_in[2];  // (128)
  const float* kernels = (const float*)d_in[3];  // (9,128)
  float* out           = (float*)d_out;          // (16,4096,9,128)

  // M = 16*4096*9 = 589824 rows -> 36864 tiles of 16 -> 8 waves/block
  const int totalTiles = (16 * 4096 * 9) / 16;   // 36864
  const int blocks     = totalTiles / 8;         // 4608
  lorentz_kernel_transform<<<blocks, 256, 0, stream>>>(x, wt, bias, kernels, out);
}